// CausalRMSNorm_80298708566411
// MI455X (gfx1250) — compile-verified
//
#include <hip/hip_runtime.h>
#include <cstdint>
#include <cstddef>

// ---------------- problem constants (match reference) ----------------
#define MOM   0.1f
#define ONE_M 0.9f
#define EPSF  1e-6f

constexpr int Bb = 4;
constexpr int Tt = 4096;
constexpr int Dd = 2048;

constexpr int TD   = 128;        // d-channels per block (== blockDim.x)
constexpr int L    = 256;        // timesteps per chunk
constexpr int C    = Tt / L;     // 16 chunks
constexpr int SUB  = 32;         // rows per TDM sub-tile
constexpr int NSUB = L / SUB;    // 8 sub-tiles, double buffered

// 0.9^256 (chunk transfer coefficient): exp(256*ln(0.9))
#define A_POW_L 1.932334e-12f

// ---------------- vector types ----------------
typedef __attribute__((ext_vector_type(4))) unsigned int u32x4;
typedef __attribute__((ext_vector_type(8))) int          i32x8;
typedef __attribute__((ext_vector_type(4))) int          i32x4;
typedef __attribute__((ext_vector_type(2))) float        v2f;
typedef __attribute__((ext_vector_type(8))) float        v8f;

// LDS aperture: flat addr low 32 bits == LDS byte offset (ISA 10.2).
__device__ __forceinline__ uint32_t lds_off32(const void* p) {
    return (uint32_t)(uintptr_t)p;
}

// ---------------- TDM (Tensor Data Mover) helper ----------------
// 2D tensor_load_to_lds: tile (tile_t rows x TD cols) of f32, row stride Dd,
// from gptr into LDS byte offset lds_off. D# layout per CDNA5 ISA.
__device__ __forceinline__ void tdm_load_tile(uint32_t lds_off, const float* gptr,
                                              int tile_t) {
    const uint64_t ga = (uint64_t)(uintptr_t)gptr;

    u32x4 g0 = {
        1u,                                                  // count=1, user desc
        lds_off,                                             // lds_addr (bytes)
        (uint32_t)(ga & 0xFFFFFFFFull),                      // global_addr[31:0]
        (uint32_t)((ga >> 32) & 0x1FFFFFFull) | (2u << 30)   // ga[56:32] | type=2
    };

    const uint32_t dim0 = (uint32_t)Dd;   // tensor_dim0
    const uint32_t dim1 = (uint32_t)Tt;   // tensor_dim1
    const uint32_t str0 = (uint32_t)Dd;   // tensor_dim0_stride (elements)

    i32x8 g1 = {
        (int)(2u << 16),                                         // data_size=4B
        (int)((dim0 & 0xFFFFu) << 16),                           // dim0[15:0]<<16
        (int)(((dim0 >> 16) & 0xFFFFu) | ((dim1 & 0xFFFFu) << 16)),
        (int)(((dim1 >> 16) & 0xFFFFu) | ((uint32_t)TD << 16)),  // tile_dim0=TD
        (int)((uint32_t)tile_t & 0xFFFFu),                       // tile_dim1, dim2=0
        (int)str0,                                               // stride0[31:0]
        0, 0
    };

    i32x4 z4 = {0, 0, 0, 0};
#if defined(__clang_major__) && (__clang_major__ >= 23)
    i32x8 z8 = {0, 0, 0, 0, 0, 0, 0, 0};
    __builtin_amdgcn_tensor_load_to_lds(g0, g1, z4, z4, z8, 0);
#else
    __builtin_amdgcn_tensor_load_to_lds(g0, g1, z4, z4, 0);
#endif
}

// ---------------- Pass 1: chunk sums via f32 WMMA over TDM-staged tiles ----
// S[c,b,d] = sum_{i=0..L-1} (MOM * ONE_M^(L-1-i)) * x[b, c*L+i, d]^2
// Weighted column reduction == chained V_WMMA_F32_16X16X4_F32:
//   A[m,k] = x^2 for 16 channels x 4 timesteps, B[k,n] = coeff (all N equal),
//   C accumulates over 64 K-steps. Two 16-channel chains per wave.
__global__ __launch_bounds__(TD)
void CausalRMSNorm_pass1_kernel(const float* __restrict__ x,
                                float* __restrict__ S_out)   // [C,B,D]
{
    __shared__ float smem[2 * SUB * TD];   // 32 KB double buffer
    __shared__ float wtab[L];              // coefficient table

    const int tid = threadIdx.x;
    const int dt  = blockIdx.x;
    const int b   = blockIdx.y;
    const int c   = blockIdx.z;
    const int d0  = dt * TD;
    const int t0  = c * L;

    const float* gbase = x + ((size_t)b * Tt + t0) * Dd + d0;
    const uint32_t lds_base = lds_off32(&smem[0]);

    // wtab[j] = MOM * ONE_M^(L-1-j)
    for (int j = tid; j < L; j += TD)
        wtab[j] = MOM * __powf(ONE_M, (float)(L - 1 - j));

    // prime the TDM pipeline (wave 0 only; TENSORcnt is per-wave)
    if (tid < 32) tdm_load_tile(lds_base, gbase, SUB);

    const int lane   = tid & 31;
    const int waveId = tid >> 5;
    const int k0     = (lane >> 4) << 1;             // 0 (lanes 0-15) / 2 (16-31)
    const int chA    = waveId * 32 + (lane & 15);    // chain A channel
    const int chB    = chA + 16;                     // chain B channel

    v8f acc0 = {0, 0, 0, 0, 0, 0, 0, 0};
    v8f acc1 = {0, 0, 0, 0, 0, 0, 0, 0};
#if !__has_builtin(__builtin_amdgcn_wmma_f32_16x16x4_f32)
    float rs_fallback = 0.0f;
#endif

    for (int s = 0; s < NSUB; ++s) {
        if (tid < 32) {
            if (s + 1 < NSUB) {
                const uint32_t dst = lds_base + (uint32_t)(((s + 1) & 1) * SUB * TD * 4);
                tdm_load_tile(dst, gbase + (size_t)(s + 1) * SUB * Dd, SUB);
                __builtin_amdgcn_s_wait_tensorcnt(1);
            } else {
                __builtin_amdgcn_s_wait_tensorcnt(0);
            }
        }
        __syncthreads();
        const float* buf = &smem[(s & 1) * SUB * TD];

#if __has_builtin(__builtin_amdgcn_wmma_f32_16x16x4_f32)
        #pragma unroll
        for (int i = 0; i < SUB; i += 4) {
            const int t = s * SUB + i;
            v2f bv = { wtab[t + k0], wtab[t + k0 + 1] };   // B: same for all N
            float x0 = buf[(i + k0) * TD + chA];
            float x1 = buf[(i + k0 + 1) * TD + chA];
            v2f av0 = { x0 * x0, x1 * x1 };
            acc0 = __builtin_amdgcn_wmma_f32_16x16x4_f32(
                false, av0, false, bv, (short)0, acc0, false, false);
            float y0 = buf[(i + k0) * TD + chB];
            float y1 = buf[(i + k0 + 1) * TD + chB];
            v2f av1 = { y0 * y0, y1 * y1 };
            acc1 = __builtin_amdgcn_wmma_f32_16x16x4_f32(
                false, av1, false, bv, (short)0, acc1, false, false);
        }
#else
        // scalar fallback: same recurrence as reference with rs0 = 0
        #pragma unroll 8
        for (int i = 0; i < SUB; ++i) {
            const float xv = buf[i * TD + tid];
            rs_fallback = __builtin_fmaf(MOM, xv * xv, ONE_M * rs_fallback);
        }
#endif
        __syncthreads();
    }

#if __has_builtin(__builtin_amdgcn_wmma_f32_16x16x4_f32)
    // C/D layout: VGPR r holds M=r (lanes 0-15) / M=8+r (lanes 16-31), N=lane%16.
    // All N columns identical -> lanes with (lane&15)==0 write 8 channels each.
    if ((lane & 15) == 0) {
        float* Sb = S_out + ((size_t)c * Bb + b) * Dd + d0;
        const int mb = waveId * 32 + ((lane >> 4) << 3);   // +0 or +8
        #pragma unroll
        for (int r = 0; r < 8; ++r) {
            Sb[mb + r]      = acc0[r];
            Sb[mb + 16 + r] = acc1[r];
        }
    }
#else
    S_out[((size_t)c * Bb + b) * Dd + d0 + tid] = rs_fallback;
#endif
}

// ---------------- Pass 3: replay chunks with carry-in, write outputs -------
__global__ __launch_bounds__(TD)
void CausalRMSNorm_pass3_kernel(const float* __restrict__ x,
                                const float* __restrict__ weight,
                                const float* __restrict__ carry_in,  // [C,B,D]
                                float* __restrict__ out)             // [B,T,D]
{
    __shared__ float smem[2 * SUB * TD];   // 32 KB double buffer

    const int tid = threadIdx.x;
    const int dt  = blockIdx.x;
    const int b   = blockIdx.y;
    const int c   = blockIdx.z;
    const int d0  = dt * TD;
    const int d   = d0 + tid;
    const int t0  = c * L;

    const float* gbase = x + ((size_t)b * Tt + t0) * Dd + d0;
    const uint32_t lds_base = lds_off32(&smem[0]);

    if (tid < 32) tdm_load_tile(lds_base, gbase, SUB);

    float rs = carry_in[((size_t)c * Bb + b) * Dd + d];
    const float w = weight[d];
    float* obase = out + ((size_t)b * Tt + t0) * Dd + d;

    for (int s = 0; s < NSUB; ++s) {
        if (tid < 32) {
            if (s + 1 < NSUB) {
                const uint32_t dst = lds_base + (uint32_t)(((s + 1) & 1) * SUB * TD * 4);
                tdm_load_tile(dst, gbase + (size_t)(s + 1) * SUB * Dd, SUB);
                __builtin_amdgcn_s_wait_tensorcnt(1);
            } else {
                __builtin_amdgcn_s_wait_tensorcnt(0);
            }
        }
        __syncthreads();

        const float* buf = &smem[(s & 1) * SUB * TD];
        #pragma unroll 8
        for (int i = 0; i < SUB; ++i) {
            const float xv = buf[i * TD + tid];   // bank-conflict-free
            rs = __builtin_fmaf(MOM, xv * xv, ONE_M * rs);
#if __has_builtin(__builtin_amdgcn_rsqf)
            const float r = __builtin_amdgcn_rsqf(rs + EPSF);
#else
            const float r = rsqrtf(rs + EPSF);
#endif
            obase[(size_t)(s * SUB + i) * Dd] = xv * r * w;
        }
        __syncthreads();
    }
}

// ---------------- Pass 2: scan the C=16 chunk carries per channel ----------
__global__ void CausalRMSNorm_chunkcarry_kernel(const float* __restrict__ running_sq, // [B,D]
                                                const float* __restrict__ S,          // [C,B,D]
                                                float* __restrict__ carry,            // [C,B,D]
                                                float* __restrict__ rs_final)         // [B,D]
{
    const int idx = blockIdx.x * blockDim.x + threadIdx.x;   // 0..B*D-1
    if (idx >= Bb * Dd) return;

    float cr = running_sq[idx];
    #pragma unroll
    for (int c = 0; c < C; ++c) {
        carry[(size_t)c * Bb * Dd + idx] = cr;               // rs entering chunk c
        cr = __builtin_fmaf(A_POW_L, cr, S[(size_t)c * Bb * Dd + idx]);
    }
    rs_final[idx] = cr;
}

// ---------------- launcher ----------------
extern "C" void kernel_launch(void* const* d_in, const int* in_sizes, int n_in,
                              void* d_out, int out_size, void* d_ws, size_t ws_size,
                              hipStream_t stream) {
    const float* x          = (const float*)d_in[0];   // [B,T,D]
    const float* running_sq = (const float*)d_in[1];   // [B,D]
    const float* weight     = (const float*)d_in[2];   // [D]

    float* out      = (float*)d_out;                   // [B,T,D]
    float* rs_final = out + (size_t)Bb * Tt * Dd;      // [B,D] (tuple order)

    float* S     = (float*)d_ws;                       // [C,B,D] 512 KB
    float* carry = S + (size_t)C * Bb * Dd;            // [C,B,D] 512 KB

    dim3 grid(Dd / TD, Bb, C);                         // 16 x 4 x 16 = 1024 blocks
    dim3 block(TD);                                    // 128 threads = 4 waves

    // Pass 1: per-chunk weighted sums via WMMA (rs0 = 0) -> S
    CausalRMSNorm_pass1_kernel<<<grid, block, 0, stream>>>(x, S);

    // Pass 2: combine chunk carries sequentially (16 steps, 8192 channels)
    CausalRMSNorm_chunkcarry_kernel<<<(Bb * Dd + 255) / 256, 256, 0, stream>>>(
        running_sq, S, carry, rs_final);

    // Pass 3: replay each chunk with correct carry-in, write outputs
    CausalRMSNorm_pass3_kernel<<<grid, block, 0, stream>>>(x, weight, carry, out);
}